// AI4MULTI_57372173140511
// MI455X (gfx1250) — compile-verified
//
#include <hip/hip_runtime.h>

// ---------------------------------------------------------------------------
// Multigrid F-cycle pressure solve, 128^3 f32, MI455X (gfx1250, wave32).
// Fine-grid 3x3x3 stencils run through V_WMMA_F32_16X16X4_F32 via a banded
// Toeplitz formulation of the x-convolution; LDS halo tiling for data reuse.
// ---------------------------------------------------------------------------

#define NF   128
#define NF3  (NF * NF * NF)
#define NP   130              // padded extent
#define DT_C 1e-4f

typedef float v2f __attribute__((ext_vector_type(2)));
typedef float v8f __attribute__((ext_vector_type(8)));

__device__ __forceinline__ int clampi(int v, int lo, int hi) {
    return v < lo ? lo : (v > hi ? hi : v);
}

// bc_pd: edge padding on all faces, then top-z padded plane (zp==129) := 0
__device__ __forceinline__ float bcpd(const float* __restrict__ s,
                                      int zp, int yp, int xp) {
    if (zp == NP - 1) return 0.0f;
    int z = clampi(zp - 1, 0, NF - 1);
    int y = clampi(yp - 1, 0, NF - 1);
    int x = clampi(xp - 1, 0, NF - 1);
    return s[(z * NF + y) * NF + x];
}

// ---------------------------------------------------------------------------
// Fused fine-grid 3x3x3 conv (WMMA core).
//   mode 0 (residual):  out = conv(bc_pd(src), wlap) - bsrc
//   mode 1 (smooth):    out = src - conv(bc_pd(src), wlap)/diag + bsrc/diag
// Block: 256 threads = 8 waves. Tile: 16(x) x 16(y) x 8(z), 1 z-plane/wave.
// ---------------------------------------------------------------------------
#define TZ   8
#define LT_Z (TZ + 2)   // 10
#define LT_Y 18
#define LT_X 20         // 18 valid + 2 zero columns so K=20 chunks are safe

__global__ __launch_bounds__(256)
void mg_conv_fine(float* __restrict__ out, const float* __restrict__ src,
                  const float* __restrict__ bsrc, const float* __restrict__ wlap,
                  int mode)
{
    __shared__ alignas(16) float Lt[LT_Z * LT_Y * LT_X];   // halo input tile
    __shared__ alignas(16) float Atab[9 * 5 * 16 * 4];     // Toeplitz A tables

    const int tid = threadIdx.x;
    const int x0 = blockIdx.x * 16;
    const int y0 = blockIdx.y * 16;
    const int z0 = blockIdx.z * TZ;

    // Load padded input tile (bc_pd applied on the fly)
    for (int i = tid; i < LT_Z * LT_Y * LT_X; i += 256) {
        int zz = i / (LT_Y * LT_X);
        int rem = i - zz * (LT_Y * LT_X);
        int yy = rem / LT_X;
        int uu = rem - yy * LT_X;
        float v = 0.0f;
        if (uu < 18) v = bcpd(src, z0 + zz, y0 + yy, x0 + uu);
        Lt[i] = v;
    }
    // Build banded Toeplitz A tables: Atab[p][c][m][k] = wx[(4c+k)-m] or 0,
    // where p = dz*3+dy indexes the 9 (dz,dy) tap pairs, wx = wlap[3p..3p+2].
    for (int i = tid; i < 9 * 5 * 64; i += 256) {
        int p = i / 320;
        int rem = i - p * 320;
        int c = rem / 64;
        int rem2 = rem - c * 64;
        int m = rem2 >> 2;
        int k = rem2 & 3;
        int t = 4 * c + k - m;
        Atab[i] = (t >= 0 && t <= 2) ? wlap[p * 3 + t] : 0.0f;
    }
    __syncthreads();

    const int lane = tid & 31;
    const int wv   = tid >> 5;    // wave id -> local z plane (0..7)
    const int h    = lane >> 4;   // K-half selector (A: K{0,1}/{2,3}; B same)
    const int nn   = lane & 15;   // N (=y) for B/D, M row for A
    const float diag = wlap[13];  // wA[0,0,1,1,1]

    v8f acc = {0.f, 0.f, 0.f, 0.f, 0.f, 0.f, 0.f, 0.f};

#if defined(__gfx1250__) && __has_builtin(__builtin_amdgcn_wmma_f32_16x16x4_f32)
    #pragma unroll
    for (int dz = 0; dz < 3; ++dz) {
        #pragma unroll
        for (int dy = 0; dy < 3; ++dy) {
            const float* Arow = Atab + (dz * 3 + dy) * 320;
            const float* Brow = Lt + ((wv + dz) * LT_Y + (nn + dy)) * LT_X;
            #pragma unroll
            for (int c = 0; c < 5; ++c) {
                // A(16x4): lane m=nn holds K = 2h, 2h+1
                v2f a  = *(const v2f*)(Arow + c * 64 + nn * 4 + 2 * h);
                // B(4x16): lane n=nn holds rows K = 2h, 2h+1 (x positions 4c+2h..)
                v2f bb = *(const v2f*)(Brow + 4 * c + 2 * h);
                acc = __builtin_amdgcn_wmma_f32_16x16x4_f32(
                    false, a, false, bb, (short)0, acc, false, false);
            }
        }
    }
#else
    // VALU fallback: identical D-layout (M = v + 8h, N = nn)
    #pragma unroll
    for (int dz = 0; dz < 3; ++dz)
        #pragma unroll
        for (int dy = 0; dy < 3; ++dy)
            #pragma unroll
            for (int dx = 0; dx < 3; ++dx) {
                float wt = wlap[dz * 9 + dy * 3 + dx];
                const float* Lr =
                    Lt + ((wv + dz) * LT_Y + (nn + dy)) * LT_X + dx;
                #pragma unroll
                for (int v = 0; v < 8; ++v)
                    acc[v] += wt * Lr[v + 8 * h];
            }
#endif

    // Epilogue: D[M=v+8h][N=nn] is the conv at (z0+wv, y0+nn, x0+M)
    const int zg = z0 + wv;
    const int yg = y0 + nn;
    const int base = (zg * NF + yg) * NF + x0;
    #pragma unroll
    for (int v = 0; v < 8; ++v) {
        int m = v + 8 * h;
        int gi = base + m;
        float bq = bsrc[gi];
        float val;
        if (mode == 0) {
            val = acc[v] - bq;
        } else {
            float center = Lt[((wv + 1) * LT_Y + (nn + 1)) * LT_X + (m + 1)];
            val = center - acc[v] / diag + bq / diag;
        }
        out[gi] = val;
    }
}

// ---------------------------------------------------------------------------
// b = (conv(uu*rhoo,w2) + conv(vv*rhoo,w3) + conv(ww*rhoo,w4)
//      + (rho - rho_old)/DT) / DT   with rhoo = bc_scalar(rho)
// ---------------------------------------------------------------------------
__global__ void mg_build_b(float* __restrict__ bout,
                           const float* __restrict__ uu,
                           const float* __restrict__ vv,
                           const float* __restrict__ ww,
                           const float* __restrict__ rho,
                           const float* __restrict__ rho_old,
                           const float* __restrict__ w2,
                           const float* __restrict__ w3,
                           const float* __restrict__ w4)
{
    int i = blockIdx.x * blockDim.x + threadIdx.x;
    if (i >= NF3) return;
    int x = i & (NF - 1);
    int y = (i >> 7) & (NF - 1);
    int z = i >> 14;
    float acc = 0.0f;
    #pragma unroll
    for (int dz = 0; dz < 3; ++dz)
        #pragma unroll
        for (int dy = 0; dy < 3; ++dy)
            #pragma unroll
            for (int dx = 0; dx < 3; ++dx) {
                int zp = z + dz, yp = y + dy, xp = x + dx;
                float rr = 0.0f;
                if (zp != 0) {  // bc_scalar: bottom-z padded plane = 0
                    int zc = clampi(zp - 1, 0, NF - 1);
                    int yc = clampi(yp - 1, 0, NF - 1);
                    int xc = clampi(xp - 1, 0, NF - 1);
                    rr = rho[(zc * NF + yc) * NF + xc];
                }
                int pidx = (zp * NP + yp) * NP + xp;
                int t = dz * 9 + dy * 3 + dx;
                acc += (uu[pidx] * w2[t] + vv[pidx] * w3[t] +
                        ww[pidx] * w4[t]) * rr;
            }
    acc += (rho[i] - rho_old[i]) / DT_C;
    bout[i] = acc / DT_C;
}

// 2x2x2 stride-2 restriction: out(s^3) from in((2s)^3)
__global__ void mg_restrict(float* __restrict__ o, const float* __restrict__ in,
                            const float* __restrict__ wres, int s)
{
    int i = blockIdx.x * blockDim.x + threadIdx.x;
    if (i >= s * s * s) return;
    int x = i % s, y = (i / s) % s, z = i / (s * s);
    int S = 2 * s;
    float acc = 0.0f;
    #pragma unroll
    for (int a = 0; a < 2; ++a)
        #pragma unroll
        for (int b = 0; b < 2; ++b)
            #pragma unroll
            for (int c = 0; c < 2; ++c)
                acc += wres[a * 4 + b * 2 + c] *
                       in[((2 * z + a) * S + 2 * y + b) * S + 2 * x + c];
    o[i] = acc;
}

// Coarse Jacobi: o = w - conv(zero_pad(w), wlap)/diag + r/diag
__global__ void mg_jacobi(float* __restrict__ o, const float* __restrict__ win,
                          const float* __restrict__ r,
                          const float* __restrict__ wlap, int s)
{
    int i = blockIdx.x * blockDim.x + threadIdx.x;
    if (i >= s * s * s) return;
    int x = i % s, y = (i / s) % s, z = i / (s * s);
    float diag = wlap[13];
    float conv = 0.0f;
    #pragma unroll
    for (int dz = 0; dz < 3; ++dz)
        #pragma unroll
        for (int dy = 0; dy < 3; ++dy)
            #pragma unroll
            for (int dx = 0; dx < 3; ++dx) {
                int zz = z + dz - 1, yy = y + dy - 1, xx = x + dx - 1;
                if (zz >= 0 && zz < s && yy >= 0 && yy < s && xx >= 0 && xx < s)
                    conv += wlap[dz * 9 + dy * 3 + dx] *
                            win[(zz * s + yy) * s + xx];
            }
    o[i] = win[i] - conv / diag + r[i] / diag;
}

// Nearest-neighbour x2 prolongation: out(s^3) from in((s/2)^3)
__global__ void mg_prolong(float* __restrict__ o, const float* __restrict__ in,
                           int s)
{
    int i = blockIdx.x * blockDim.x + threadIdx.x;
    if (i >= s * s * s) return;
    int x = i % s, y = (i / s) % s, z = i / (s * s);
    int hs = s >> 1;
    o[i] = in[(((z >> 1) * hs) + (y >> 1)) * hs + (x >> 1)];
}

__global__ void mg_copy(float* __restrict__ o, const float* __restrict__ a, int n)
{
    int i = blockIdx.x * blockDim.x + threadIdx.x;
    if (i < n) o[i] = a[i];
}

__global__ void mg_zero(float* __restrict__ o, int n)
{
    int i = blockIdx.x * blockDim.x + threadIdx.x;
    if (i < n) o[i] = 0.0f;
}

__global__ void mg_sub(float* __restrict__ o, const float* __restrict__ a,
                       const float* __restrict__ b, int n)
{
    int i = blockIdx.x * blockDim.x + threadIdx.x;
    if (i < n) o[i] = a[i] - b[i];
}

// ---------------------------------------------------------------------------
extern "C" void kernel_launch(void* const* d_in, const int* in_sizes, int n_in,
                              void* d_out, int out_size, void* d_ws, size_t ws_size,
                              hipStream_t stream)
{
    const float* uu      = (const float*)d_in[0];
    const float* vv      = (const float*)d_in[1];
    const float* ww      = (const float*)d_in[2];
    const float* rho     = (const float*)d_in[3];
    const float* rho_old = (const float*)d_in[4];
    const float* pd_in   = (const float*)d_in[5];
    const float* w2      = (const float*)d_in[6];
    const float* w3      = (const float*)d_in[7];
    const float* w4      = (const float*)d_in[8];
    const float* wlap    = (const float*)d_in[9];
    const float* wres    = (const float*)d_in[10];

    float* P0 = (float*)d_out;       // current values_pd
    float* ws = (float*)d_ws;
    size_t off = 0;
    float* b  = ws + off; off += NF3;
    float* r0 = ws + off; off += NF3;
    float* r1 = ws + off; off += 64 * 64 * 64;
    float* r2 = ws + off; off += 32 * 32 * 32;
    float* r3 = ws + off; off += 16 * 16 * 16;
    float* r4 = ws + off; off += 8 * 8 * 8;
    float* wa = ws + off; off += NF3;           // correction w (up to 128^3)
    float* wb = ws + off; off += 64 * 64 * 64;  // jacobi temp (<= 64^3)
    float* P1 = ws + off; off += NF3;           // pd ping-pong

    const int TPB = 256;
    auto g1 = [](int n) { return dim3((unsigned)((n + 255) / 256)); };
    dim3 cgrid(NF / 16, NF / 16, NF / TZ);

    mg_copy   <<<g1(NF3), TPB, 0, stream>>>(P0, pd_in, NF3);
    mg_build_b<<<g1(NF3), TPB, 0, stream>>>(b, uu, vv, ww, rho, rho_old,
                                            w2, w3, w4);

    for (int it = 0; it < 5; ++it) {
        // fine residual
        mg_conv_fine<<<cgrid, 256, 0, stream>>>(r0, P0, b, wlap, 0);
        // restrict down: 128 -> 64 -> 32 -> 16 -> 8
        mg_restrict<<<g1(64*64*64), TPB, 0, stream>>>(r1, r0, wres, 64);
        mg_restrict<<<g1(32*32*32), TPB, 0, stream>>>(r2, r1, wres, 32);
        mg_restrict<<<g1(16*16*16), TPB, 0, stream>>>(r3, r2, wres, 16);
        mg_restrict<<<g1(8*8*8),    TPB, 0, stream>>>(r4, r3, wres, 8);
        // Jacobi-correct + prolong up: 8 -> 16 -> 32 -> 64 -> 128
        mg_zero   <<<g1(512),    TPB, 0, stream>>>(wa, 512);
        mg_jacobi <<<g1(512),    TPB, 0, stream>>>(wb, wa, r4, wlap, 8);
        mg_prolong<<<g1(4096),   TPB, 0, stream>>>(wa, wb, 16);
        mg_jacobi <<<g1(4096),   TPB, 0, stream>>>(wb, wa, r3, wlap, 16);
        mg_prolong<<<g1(32768),  TPB, 0, stream>>>(wa, wb, 32);
        mg_jacobi <<<g1(32768),  TPB, 0, stream>>>(wb, wa, r2, wlap, 32);
        mg_prolong<<<g1(262144), TPB, 0, stream>>>(wa, wb, 64);
        mg_jacobi <<<g1(262144), TPB, 0, stream>>>(wb, wa, r1, wlap, 64);
        mg_prolong<<<g1(NF3),    TPB, 0, stream>>>(wa, wb, 128);
        // apply correction + one fine Jacobi smoothing sweep
        mg_sub      <<<g1(NF3), TPB, 0, stream>>>(P1, P0, wa, NF3);
        mg_conv_fine<<<cgrid, 256, 0, stream>>>(P0, P1, b, wlap, 1);
    }
    (void)in_sizes; (void)n_in; (void)out_size; (void)ws_size;
}